// HierarchicalBayesianLinear_49022756716597
// MI455X (gfx1250) — compile-verified
//
#include <hip/hip_runtime.h>
#include <hip/hip_bf16.h>

// Problem constants
#define BN 4096
#define IN 256
#define ON 256
#define GN 32

typedef float v2f __attribute__((ext_vector_type(2)));
typedef float v8f __attribute__((ext_vector_type(8)));

__device__ __forceinline__ float softplus_f(float x) {
    // numerically stable log1p(exp(x))
    return (x > 0.0f) ? x + log1pf(expf(-x)) : log1pf(expf(x));
}

// ---------------------------------------------------------------------------
// pre0: tiny tables: gws[o*I+i]=sp(gw_rho), ard_scale[i], gbs[o], rbs[g*O+o]
// grid: (O*I/256) blocks x 256
// ---------------------------------------------------------------------------
__global__ void k_pre0(const float* __restrict__ gw_rho,
                       const float* __restrict__ gb_rho,
                       const float* __restrict__ rb_rho,
                       const float* __restrict__ ard_a,
                       const float* __restrict__ ard_b,
                       float* __restrict__ gws,
                       float* __restrict__ gbs,
                       float* __restrict__ rbs,
                       float* __restrict__ ard_scale) {
    int idx = blockIdx.x * blockDim.x + threadIdx.x;
    if (idx < ON * IN) gws[idx] = softplus_f(gw_rho[idx]);
    if (idx < GN * ON) rbs[idx] = softplus_f(rb_rho[idx]);
    if (idx < ON)      gbs[idx] = softplus_f(gb_rho[idx]);
    if (idx < IN)      ard_scale[idx] = softplus_f(ard_a[idx]) * softplus_f(ard_b[idx]);
}

// ---------------------------------------------------------------------------
// pre1: CT[g,o,i] = {gwσ·rwμ, gwσ·rwσ} interleaved float2 ; xs = x·ard_scale
// grid: (G*O*I/256) blocks x 256
// ---------------------------------------------------------------------------
__global__ void k_pre1(const float* __restrict__ x,
                       const float* __restrict__ rw_mu,
                       const float* __restrict__ rw_rho,
                       const float* __restrict__ gws,
                       const float* __restrict__ ard_scale,
                       float* __restrict__ xs,
                       float2* __restrict__ ct) {
    int idx = blockIdx.x * blockDim.x + threadIdx.x;
    if (idx < GN * ON * IN) {
        int rem = idx & (ON * IN - 1);          // (o*I + i)
        float gs = gws[rem];
        float2 v;
        v.x = gs * rw_mu[idx];
        v.y = gs * softplus_f(rw_rho[idx]);
        ct[idx] = v;
    }
    if (idx < BN * IN) {
        xs[idx] = x[idx] * ard_scale[idx & (IN - 1)];
    }
}

// ---------------------------------------------------------------------------
// GEMM term: out[b,o] = sum_i xs[b,i] * gw_mu[o,i]  via V_WMMA_F32_16X16X4_F32
// One wave computes a 16x16 C tile, K-loop over 256 in steps of 4.
// grid: 512 blocks x 256 threads (8 waves/block, 4096 tiles)
// ---------------------------------------------------------------------------
__global__ void k_gemm_wmma(const float* __restrict__ xs,
                            const float* __restrict__ gmu,
                            float* __restrict__ out) {
    int wid  = (blockIdx.x * blockDim.x + threadIdx.x) >> 5;  // 0..4095
    int lane = threadIdx.x & 31;
    int tb = wid >> 4;          // B tile: 0..255
    int to = wid & 15;          // O tile: 0..15
    int b0 = tb << 4;
    int o0 = to << 4;
    int m  = lane & 15;         // row (A) / col (B)
    int kh = lane >> 4;         // K half: 0 or 1 (covers K pairs {0,1} / {2,3})

    const float* arow = xs  + (size_t)(b0 + m) * IN;
    const float* brow = gmu + (size_t)(o0 + m) * IN;

    v8f c = {};
    for (int k = 0; k < IN; k += 4) {
        v2f a = *(const v2f*)(arow + k + 2 * kh);
        v2f b = *(const v2f*)(brow + k + 2 * kh);
        c = __builtin_amdgcn_wmma_f32_16x16x4_f32(
                false, a, false, b, (short)0, c, false, false);
    }

    // C layout: VGPR r -> lanes0-15: M=r ; lanes16-31: M=r+8 ; N = lane&15
#pragma unroll
    for (int r = 0; r < 8; ++r) {
        int row = b0 + r + (kh << 3);
        out[(size_t)row * ON + o0 + m] = c[r];
    }
}

// ---------------------------------------------------------------------------
// Streaming kernel (HBM-bound): out[b,o] += Σ_i xs[b,i]·(M2 + T·eps_w[b,o,i])
//                               + bias[b,o]
// One block per b; 8 waves x 32 o-columns each; b128 coalesced loads.
// ---------------------------------------------------------------------------
__global__ void k_main(const float* __restrict__ xs,
                       const float2* __restrict__ ct,
                       const float* __restrict__ gbs,
                       const float* __restrict__ rbs,
                       const float* __restrict__ gb_mu,
                       const float* __restrict__ rb_mu,
                       const float* __restrict__ eps_w,
                       const float* __restrict__ eps_b,
                       const int* __restrict__ gid,
                       float* __restrict__ out) {
    __shared__ float4 xls4[IN / 4];
    float* xls = (float*)xls4;

    int b   = blockIdx.x;
    int tid = threadIdx.x;
    xls[tid] = xs[(size_t)b * IN + tid];
    __syncthreads();

    int g    = gid[b];
    int wave = tid >> 5;
    int lane = tid & 31;

    const float2* ctg  = ct + (size_t)g * ON * IN;
    const float*  epsb = eps_w + (size_t)b * ON * IN;

    for (int oo = 0; oo < 32; ++oo) {
        int o = (wave << 5) + oo;
        const float4* e4 = (const float4*)(epsb + (size_t)o * IN);
        const float4* c4 = (const float4*)(ctg + (size_t)o * IN);  // pairs

        float acc = 0.0f;
#pragma unroll
        for (int cchunk = 0; cchunk < 2; ++cchunk) {
            int base = (cchunk << 7) + (lane << 2);   // 4*lane + 128*c
            float4 ev = e4[base >> 2];
            float4 ca = c4[base >> 1];                // ct pairs base, base+1
            float4 cb = c4[(base >> 1) + 1];          // ct pairs base+2, base+3
            float4 xv = xls4[base >> 2];
            acc = fmaf(xv.x, fmaf(ca.y, ev.x, ca.x), acc);
            acc = fmaf(xv.y, fmaf(ca.w, ev.y, ca.z), acc);
            acc = fmaf(xv.z, fmaf(cb.y, ev.z, cb.x), acc);
            acc = fmaf(xv.w, fmaf(cb.w, ev.w, cb.z), acc);
        }
#pragma unroll
        for (int off = 16; off > 0; off >>= 1)
            acc += __shfl_xor(acc, off, 32);

        if (lane == 0) {
            size_t bo = (size_t)b * ON + o;
            int go = g * ON + o;
            float biasv = gb_mu[o] + gbs[o] * fmaf(rbs[go], eps_b[bo], rb_mu[go]);
            out[bo] += acc + biasv;
        }
    }
}

// ---------------------------------------------------------------------------
// KL partial reduction over G*O*I domain (first O*I / O / I lanes also fold
// in the group / bias / ARD terms). Deterministic: per-block partials.
// ---------------------------------------------------------------------------
__global__ void k_kl_part(const float* __restrict__ gw_mu,
                          const float* __restrict__ gw_rho,
                          const float* __restrict__ rw_mu,
                          const float* __restrict__ rw_rho,
                          const float* __restrict__ gws,
                          const float* __restrict__ gbs,
                          const float* __restrict__ ard_a,
                          const float* __restrict__ ard_b,
                          float* __restrict__ partials) {
    const float c_half = -0.9189385332046727f;  // -0.5*log(2*pi)
    int idx = blockIdx.x * blockDim.x + threadIdx.x;
    float local = 0.0f;

    if (idx < GN * ON * IN) {   // individual-level KL
        float rs  = softplus_f(rw_rho[idx]);
        float rmu = rw_mu[idx];
        local += c_half - logf(rs) + 0.5f * (rs * rs + rmu * rmu) - 0.5f;
    }
    if (idx < ON * IN) {        // group-level KL + HalfNormal(gw_sigma)
        float gs  = gws[idx];
        float gmu = gw_mu[idx];
        float ls  = logf(gs);
        local += -ls + 0.5f * (gs * gs + gmu * gmu) - 0.5f;
        local += c_half - ls + 0.5f * (ls * ls + gs * gs) - 0.5f;
    }
    if (idx < ON) {             // HalfNormal(gb_sigma)
        float s  = gbs[idx];
        float ls = logf(s);
        local += c_half - ls + 0.5f * (ls * ls + s * s) - 0.5f;
    }
    if (idx < IN) {             // ARD terms
        local += softplus_f(ard_a[idx]) + softplus_f(ard_b[idx]);
    }

    __shared__ float red[256];
    red[threadIdx.x] = local;
    __syncthreads();
#pragma unroll
    for (int s = 128; s > 0; s >>= 1) {
        if (threadIdx.x < s) red[threadIdx.x] += red[threadIdx.x + s];
        __syncthreads();
    }
    if (threadIdx.x == 0) partials[blockIdx.x] = red[0];
}

__global__ void k_kl_finish(const float* __restrict__ partials,
                            float* __restrict__ kl_out) {
    __shared__ float red[256];
    float local = 0.0f;
    for (int i = threadIdx.x; i < 8192; i += 256) local += partials[i];
    red[threadIdx.x] = local;
    __syncthreads();
#pragma unroll
    for (int s = 128; s > 0; s >>= 1) {
        if (threadIdx.x < s) red[threadIdx.x] += red[threadIdx.x + s];
        __syncthreads();
    }
    if (threadIdx.x == 0) *kl_out = red[0];
}

// ---------------------------------------------------------------------------
extern "C" void kernel_launch(void* const* d_in, const int* in_sizes, int n_in,
                              void* d_out, int out_size, void* d_ws, size_t ws_size,
                              hipStream_t stream) {
    (void)in_sizes; (void)n_in; (void)out_size; (void)ws_size;

    const float* x      = (const float*)d_in[0];
    const float* gw_mu  = (const float*)d_in[1];
    const float* gw_rho = (const float*)d_in[2];
    const float* gb_mu  = (const float*)d_in[3];
    const float* gb_rho = (const float*)d_in[4];
    const float* rw_mu  = (const float*)d_in[5];
    const float* rw_rho = (const float*)d_in[6];
    const float* rb_mu  = (const float*)d_in[7];
    const float* rb_rho = (const float*)d_in[8];
    const float* ard_a  = (const float*)d_in[9];
    const float* ard_b  = (const float*)d_in[10];
    const float* eps_w  = (const float*)d_in[11];
    const float* eps_b  = (const float*)d_in[12];
    const int*   gid    = (const int*)d_in[13];

    float* out = (float*)d_out;                 // [B*O] then kl at [B*O]

    // workspace layout (floats)
    float* ws        = (float*)d_ws;
    float* xs        = ws;                                   // B*I   = 1048576
    float* ct        = xs + (size_t)BN * IN;                 // 2*G*O*I = 4194304
    float* gws       = ct + (size_t)2 * GN * ON * IN;        // O*I   = 65536
    float* gbs       = gws + ON * IN;                        // O     = 256
    float* rbs       = gbs + ON;                             // G*O   = 8192
    float* ard_scale = rbs + GN * ON;                        // I     = 256
    float* partials  = ard_scale + IN;                       // 8192

    dim3 blk(256);

    k_pre0<<<dim3((ON * IN) / 256), blk, 0, stream>>>(
        gw_rho, gb_rho, rb_rho, ard_a, ard_b, gws, gbs, rbs, ard_scale);

    k_pre1<<<dim3((GN * ON * IN) / 256), blk, 0, stream>>>(
        x, rw_mu, rw_rho, gws, ard_scale, xs, (float2*)ct);

    k_gemm_wmma<<<dim3((BN / 16) * (ON / 16) * 32 / 256), blk, 0, stream>>>(
        xs, gw_mu, out);

    k_main<<<dim3(BN), blk, 0, stream>>>(
        xs, (const float2*)ct, gbs, rbs, gb_mu, rb_mu, eps_w, eps_b, gid, out);

    k_kl_part<<<dim3((GN * ON * IN) / 256), blk, 0, stream>>>(
        gw_mu, gw_rho, rw_mu, rw_rho, gws, gbs, ard_a, ard_b, partials);

    k_kl_finish<<<dim3(1), blk, 0, stream>>>(partials, out + (size_t)BN * ON);
}